// Mash_2671469658781
// MI455X (gfx1250) — compile-verified
//
#include <hip/hip_runtime.h>

typedef __attribute__((ext_vector_type(2))) float v2f;
typedef __attribute__((ext_vector_type(4))) float v4f;
typedef __attribute__((ext_vector_type(8))) float v8f;

#define FFT_SZ   256
#define N_OFDM   14
#define N_EFF    216
#define M_ROWS   256            // B*NTX*NSTR = 32*4*2
#define N_FULL   3024           // N_OFDM * N_EFF
#define KC       16             // K chunk per LDS stage (4 WMMA K-steps)
#define MBLK     64             // M rows per block (4 waves x 16)
#define TWO_PI_F 6.283185307179586f

// ---------------------------------------------------------------------------
// Gather: x[B,T,S,O,256] complex -> Xre/Xim planes [256 x 3024] f32
// ---------------------------------------------------------------------------
__global__ __launch_bounds__(256) void gather_kernel(
    const float* __restrict__ x, const long long* __restrict__ sc,
    float* __restrict__ Xre, float* __restrict__ Xim) {
  int idx = blockIdx.x * blockDim.x + threadIdx.x;
  if (idx >= M_ROWS * N_FULL) return;
  int row = idx / N_FULL;
  int c   = idx - row * N_FULL;
  int o   = c / N_EFF;
  int e   = c - o * N_EFF;
  long long f  = sc[e];
  long long ci = ((long long)row * N_OFDM + o) * FFT_SZ + f;  // complex index
  Xre[idx] = x[ci * 2 + 0];
  Xim[idx] = x[ci * 2 + 1];
}

// ---------------------------------------------------------------------------
// One hop: Y[m,j] = diag[j] * sum_k X[m,k] * exp(-2*pi*i*k*j/n)/sqrt(n)
// Complex GEMM via V_WMMA_F32_16X16X4_F32; F tiles synthesized on the fly.
// Block: 128 threads = 4 wave32; 64(M) x 16(N) output per block.
// ---------------------------------------------------------------------------
__global__ __launch_bounds__(128) void dft_gemm_kernel(
    const float* __restrict__ Xre, const float* __restrict__ Xim,
    const float* __restrict__ diag,   // [N_FULL] complex interleaved
    float* __restrict__ Yre, float* __restrict__ Yim) {
  __shared__ float sAre[MBLK * KC];
  __shared__ float sAim[MBLK * KC];
  __shared__ float sBre[KC * 16];
  __shared__ float sBim[KC * 16];

  const int tid  = threadIdx.x;
  const int wave = tid >> 5;
  const int lane = tid & 31;
  const int m0   = blockIdx.y * MBLK;
  const int j0   = blockIdx.x * 16;
  const float scale = rsqrtf((float)N_FULL);

  const int mIdx = lane & 15;      // row within 16x16 fragment (A) / col (B,C)
  const int half = lane >> 4;      // lane group
  // cooperative A-tile load mapping: 64 rows x 16 cols, 2 threads/row
  const int arow = tid >> 1;
  const int acol = (tid & 1) * 8;

  v8f accRe = {};
  v8f accIm = {};

  for (int kc = 0; kc < N_FULL; kc += KC) {
    __syncthreads();  // previous iteration's fragment reads done

    // ---- stage A panel (64 x 16 re + im) into LDS, coalesced float4s ----
    const float* gre = Xre + (size_t)(m0 + arow) * N_FULL + kc + acol;
    const float* gim = Xim + (size_t)(m0 + arow) * N_FULL + kc + acol;
    v4f r0 = *(const v4f*)(gre);
    v4f r1 = *(const v4f*)(gre + 4);
    v4f i0 = *(const v4f*)(gim);
    v4f i1 = *(const v4f*)(gim + 4);
    *(v4f*)&sAre[arow * KC + acol + 0] = r0;
    *(v4f*)&sAre[arow * KC + acol + 4] = r1;
    *(v4f*)&sAim[arow * KC + acol + 0] = i0;
    *(v4f*)&sAim[arow * KC + acol + 4] = i1;
    if (kc + KC < N_FULL) {                 // pull next K chunk toward L2/L0
      __builtin_prefetch(gre + KC, 0, 1);
      __builtin_prefetch(gim + KC, 0, 1);
    }

    // ---- synthesize B tile: F[k,j] = exp(-2*pi*i*(k*j mod n)/n)/sqrt(n) ----
    #pragma unroll
    for (int q = tid; q < KC * 16; q += 128) {
      int kl = q >> 4, jl = q & 15;
      int gk = kc + kl, gj = j0 + jl;
      int ph = (int)(((long long)gk * (long long)gj) % N_FULL);  // exact phase
      float ang = -TWO_PI_F * (float)ph * (1.0f / (float)N_FULL);
      float sv, cv;
      __sincosf(ang, &sv, &cv);
      sBre[q] = cv * scale;
      sBim[q] = sv * scale;
    }
    __syncthreads();

    // ---- 4 WMMA K-steps of K=4 each; 4 real WMMAs per step (complex) ----
    #pragma unroll
    for (int ks = 0; ks < 4; ++ks) {
      const int kb    = ks * 4 + 2 * half;         // per-lane K base
      const int abase = (wave * 16 + mIdx) * KC + kb;
      v2f are, aim, bre, bim;
      are.x = sAre[abase];     are.y = sAre[abase + 1];
      aim.x = sAim[abase];     aim.y = sAim[abase + 1];
      bre.x = sBre[(kb + 0) * 16 + mIdx];
      bre.y = sBre[(kb + 1) * 16 + mIdx];
      bim.x = sBim[(kb + 0) * 16 + mIdx];
      bim.y = sBim[(kb + 1) * 16 + mIdx];
      v2f aimn = -aim;  // VALU negate (F32 WMMA NEG bits only cover C)

      accRe = __builtin_amdgcn_wmma_f32_16x16x4_f32(
          false, are,  false, bre, (short)0, accRe, false, false);
      accRe = __builtin_amdgcn_wmma_f32_16x16x4_f32(
          false, aimn, false, bim, (short)0, accRe, false, false);
      accIm = __builtin_amdgcn_wmma_f32_16x16x4_f32(
          false, are,  false, bim, (short)0, accIm, false, false);
      accIm = __builtin_amdgcn_wmma_f32_16x16x4_f32(
          false, aim,  false, bre, (short)0, accIm, false, false);
    }
  }

  // ---- epilogue: fused complex diagonal scale + store ----
  const int col = j0 + mIdx;
  const float dre = diag[col * 2 + 0];
  const float dim = diag[col * 2 + 1];
  const int rowBase = m0 + wave * 16 + 8 * half;
  #pragma unroll
  for (int r = 0; r < 8; ++r) {
    float yr = accRe[r], yi = accIm[r];
    Yre[(size_t)(rowBase + r) * N_FULL + col] = yr * dre - yi * dim;
    Yim[(size_t)(rowBase + r) * N_FULL + col] = yr * dim + yi * dre;
  }
}

// ---------------------------------------------------------------------------
// Scatter: Y planes [256 x 3024] -> out [256,14,256] complex, zero guards
// ---------------------------------------------------------------------------
__global__ __launch_bounds__(256) void scatter_kernel(
    const float* __restrict__ Yre, const float* __restrict__ Yim,
    const long long* __restrict__ sc, float* __restrict__ out) {
  int idx = blockIdx.x * blockDim.x + threadIdx.x;
  if (idx >= M_ROWS * N_OFDM * FFT_SZ) return;
  int f  = idx % FFT_SZ;
  int ro = idx / FFT_SZ;                 // row*14 + o
  int base = (int)sc[0];                 // contiguous effective band
  int e = f - base;
  float re = 0.0f, im = 0.0f;
  if (e >= 0 && e < N_EFF) {
    int row = ro / N_OFDM;
    int o   = ro - row * N_OFDM;
    size_t yi = (size_t)row * N_FULL + o * N_EFF + e;
    re = Yre[yi];
    im = Yim[yi];
  }
  out[(size_t)idx * 2 + 0] = re;
  out[(size_t)idx * 2 + 1] = im;
}

// ---------------------------------------------------------------------------
extern "C" void kernel_launch(void* const* d_in, const int* in_sizes, int n_in,
                              void* d_out, int out_size, void* d_ws, size_t ws_size,
                              hipStream_t stream) {
  const float*     x    = (const float*)d_in[0];      // complex64 interleaved
  const float*     diag = (const float*)d_in[1];      // [3, 3024] complex64
  const long long* sc   = (const long long*)d_in[2];  // [216] int64

  const size_t plane = (size_t)M_ROWS * N_FULL;       // 774,144 floats
  float* ws  = (float*)d_ws;                          // needs 4 planes ~12.4 MB
  float* Are = ws + 0 * plane;
  float* Aim = ws + 1 * plane;
  float* Bre = ws + 2 * plane;
  float* Bim = ws + 3 * plane;

  gather_kernel<<<(int)((plane + 255) / 256), 256, 0, stream>>>(x, sc, Are, Aim);

  float* inRe = Are; float* inIm = Aim;
  float* otRe = Bre; float* otIm = Bim;
  dim3 grid(N_FULL / 16, M_ROWS / MBLK);              // 189 x 4 blocks
  for (int it = 0; it < 3; ++it) {
    dft_gemm_kernel<<<grid, 128, 0, stream>>>(
        inRe, inIm, diag + (size_t)it * N_FULL * 2, otRe, otIm);
    float* t;
    t = inRe; inRe = otRe; otRe = t;
    t = inIm; inIm = otIm; otIm = t;
  }

  int total = M_ROWS * N_OFDM * FFT_SZ;
  scatter_kernel<<<(total + 255) / 256, 256, 0, stream>>>(inRe, inIm, sc,
                                                          (float*)d_out);
}